// AdaptiveMixGNNLayer_10213432229976
// MI455X (gfx1250) — compile-verified
//
#include <hip/hip_runtime.h>

#define N_NODES 50000
#define N_EDGES 800000
#define D 128
#define MTILES (N_NODES / 16)   // 3125, exact

typedef __attribute__((ext_vector_type(2))) float v2f;
typedef __attribute__((ext_vector_type(8))) float v8f;

// ---------------------------------------------------------------------------
// Zero the z_mix accumulator workspace.
// ---------------------------------------------------------------------------
__global__ __launch_bounds__(256) void zero_ws_kernel(float* __restrict__ p, int n) {
    int i = blockIdx.x * blockDim.x + threadIdx.x;
    if (i < n) p[i] = 0.0f;
}

// ---------------------------------------------------------------------------
// alpha[i] = sigmoid(x[i] . alpha_w + alpha_b)   (one wave32 per node)
// Writes to both the workspace copy (consumed by SpMM) and d_out tail.
// ---------------------------------------------------------------------------
__global__ __launch_bounds__(256) void alpha_kernel(const float* __restrict__ x,
                                                    const float* __restrict__ aw,
                                                    const float* __restrict__ ab,
                                                    float* __restrict__ alpha_ws,
                                                    float* __restrict__ alpha_out) {
    int wave = (blockIdx.x * blockDim.x + threadIdx.x) >> 5;
    int lane = threadIdx.x & 31;
    if (wave >= N_NODES) return;
    const float* xr = x + (size_t)wave * D;
    float s = 0.0f;
#pragma unroll
    for (int j = 0; j < 4; ++j) {
        int f = lane + 32 * j;
        s += xr[f] * aw[f];
    }
    // wave32 butterfly reduction
#pragma unroll
    for (int off = 16; off > 0; off >>= 1)
        s += __shfl_xor(s, off, 32);
    if (lane == 0) {
        float a = 1.0f / (1.0f + __expf(-(s + ab[0])));
        alpha_ws[wave]  = a;
        alpha_out[wave] = a;
    }
}

// ---------------------------------------------------------------------------
// Fused SpMM for both operators with the alpha gate folded into the edge
// weight:  z_mix[r] += (gate * val) * x[c].   One wave32 per edge; each lane
// handles 4 of the 128 features via global_atomic_add_f32.
// ---------------------------------------------------------------------------
__global__ __launch_bounds__(256) void spmm_mix_kernel(const float* __restrict__ x,
                                                       const int*   __restrict__ lp_rows,
                                                       const int*   __restrict__ lp_cols,
                                                       const float* __restrict__ lp_vals,
                                                       const int*   __restrict__ hp_rows,
                                                       const int*   __restrict__ hp_cols,
                                                       const float* __restrict__ hp_vals,
                                                       const float* __restrict__ alpha,
                                                       float*       __restrict__ zmix) {
    int gw   = (blockIdx.x * blockDim.x + threadIdx.x) >> 5;
    int lane = threadIdx.x & 31;
    if (gw >= 2 * N_EDGES) return;

    int r, c;
    float scale;
    if (gw < N_EDGES) {                       // low-pass operator
        int e = gw;
        r = lp_rows[e]; c = lp_cols[e];
        scale = alpha[r] * lp_vals[e];
    } else {                                  // high-pass operator
        int e = gw - N_EDGES;
        r = hp_rows[e]; c = hp_cols[e];
        scale = (1.0f - alpha[r]) * hp_vals[e];
    }

    const float* xc = x    + (size_t)c * D;
    float*       zr = zmix + (size_t)r * D;
#pragma unroll
    for (int j = 0; j < 4; ++j) {
        int f = lane + 32 * j;
        atomicAdd(&zr[f], scale * xc[f]);
    }
}

// ---------------------------------------------------------------------------
// out = relu(z_mix @ W^T + b) via V_WMMA_F32_16X16X4_F32 (full-precision f32
// matrix pipe).  Block = 256 threads = 8 waves; wave w owns N-tile w, block
// owns one M-tile of 16 rows.  K=128 -> 32 chained WMMAs per tile.
//
// A (16x4 f32) lane layout (ISA 7.12.2): lanes 0-15 hold rows M=lane with
// K = k,k+1 in VGPR0/1; lanes 16-31 hold the same rows with K = k+2,k+3.
// B (4x16) mirrors the K split with N striped across lanes.
// C/D (16x16 f32): VGPR i -> M = m0+i (lanes 0-15) / m0+i+8 (lanes 16-31),
// N = n0 + (lane & 15).
// ---------------------------------------------------------------------------
__global__ __launch_bounds__(256) void gemm_relu_kernel(const float* __restrict__ zmix,
                                                        const float* __restrict__ W,
                                                        const float* __restrict__ bias,
                                                        float*       __restrict__ out) {
    int wave = threadIdx.x >> 5;
    int lane = threadIdx.x & 31;
    int m0   = blockIdx.x * 16;
    int n0   = wave * 16;

    int half = lane >> 4;       // 0: K pair (k,k+1), 1: K pair (k+2,k+3)
    int l15  = lane & 15;

    const float* arp = zmix + (size_t)(m0 + l15) * D + 2 * half;  // A: row of z_mix
    const float* bcp = W    + (size_t)(n0 + l15) * D + 2 * half;  // B: B[k][n]=W[n][k]

    v8f c = {};  // f32 accumulator, 8 VGPRs
#pragma unroll
    for (int k = 0; k < D; k += 4) {
        v2f a = *(const v2f*)(arp + k);
        v2f b = *(const v2f*)(bcp + k);
        // (neg_a, A, neg_b, B, c_mod, C, reuse_a, reuse_b)
        c = __builtin_amdgcn_wmma_f32_16x16x4_f32(false, a, false, b,
                                                  (short)0, c, false, false);
    }

    int n = n0 + l15;
    float bn = bias[n];
#pragma unroll
    for (int i = 0; i < 8; ++i) {
        int m = m0 + i + 8 * half;
        float v = c[i] + bn;
        out[(size_t)m * D + n] = fmaxf(v, 0.0f);
    }
}

// ---------------------------------------------------------------------------
extern "C" void kernel_launch(void* const* d_in, const int* in_sizes, int n_in,
                              void* d_out, int out_size, void* d_ws, size_t ws_size,
                              hipStream_t stream) {
    const float* x       = (const float*)d_in[0];
    const int*   lp_rows = (const int*)  d_in[1];
    const int*   lp_cols = (const int*)  d_in[2];
    const float* lp_vals = (const float*)d_in[3];
    const int*   hp_rows = (const int*)  d_in[4];
    const int*   hp_cols = (const int*)  d_in[5];
    const float* hp_vals = (const float*)d_in[6];
    const float* alpha_w = (const float*)d_in[7];
    const float* alpha_b = (const float*)d_in[8];
    const float* W       = (const float*)d_in[9];
    const float* b       = (const float*)d_in[10];

    float* out       = (float*)d_out;                  // [N_NODES, D]
    float* alpha_out = out + (size_t)N_NODES * D;      // [N_NODES]  (tuple tail)

    float* zmix     = (float*)d_ws;                    // [N_NODES, D] accumulator
    float* alpha_ws = zmix + (size_t)N_NODES * D;      // [N_NODES]

    // 1) zero the z_mix accumulator
    {
        int n = N_NODES * D;
        zero_ws_kernel<<<(n + 255) / 256, 256, 0, stream>>>(zmix, n);
    }
    // 2) per-node gate alpha
    {
        int waves  = N_NODES;
        int blocks = (waves + 7) / 8;                  // 8 waves per 256-thread block
        alpha_kernel<<<blocks, 256, 0, stream>>>(x, alpha_w, alpha_b, alpha_ws, alpha_out);
    }
    // 3) fused gated SpMM over both operators (2 * 800k edges, 1 wave each)
    {
        int waves  = 2 * N_EDGES;
        int blocks = (waves + 7) / 8;
        spmm_mix_kernel<<<blocks, 256, 0, stream>>>(x, lp_rows, lp_cols, lp_vals,
                                                    hp_rows, hp_cols, hp_vals,
                                                    alpha_ws, zmix);
    }
    // 4) dense projection + bias + ReLU on the f32 WMMA pipe
    {
        gemm_relu_kernel<<<MTILES, 256, 0, stream>>>(zmix, W, b, out);
    }
}